// DiffLinearAttention_18975165514223
// MI455X (gfx1250) — compile-verified
//
#include <hip/hip_runtime.h>
#include <hip/hip_bf16.h>

// ----------------------------------------------------------------------------
// Differential quadratic causal linear attention for MI455X (gfx1250, wave32).
//
//   s = q1 k1^T - alpha * q2 k2^T  ==  [q1|q2] @ [k1 | -alpha*k2]^T
//
// Phase 1 (dla_fmap_kernel): per-head feature maps via V_WMMA_F32_16X16X4_F32,
//   writes 128-wide concatenated QC / KC into d_ws (fp32: the signed,
//   cancelling row-sum denominator makes low precision risky).
// Phase 2 (dla_attn_kernel): 4 waves per block, each owning one 16-row tile of
//   the same head. KC (8KB) and V (4KB) column tiles are staged in LDS with
//   double-buffered GLOBAL_LOAD_ASYNC_TO_LDS_B128 (ASYNCcnt) and shared by all
//   4 waves -> 4x less L2 traffic on the score side. Scores via a 32-step
//   f32 WMMA chain (K=128), diagonal causal mask via v_cndmask (EXEC
//   preserved), per-wave LDS transpose into A-layout, 16 WMMAs for s@V,
//   deferred normalization out = acc / (rowsum + eps).
//
// Workspace requirement: 2 * 64 * 1024 * 128 * 4B = 67.1 MB (QC + KC).
// ----------------------------------------------------------------------------

typedef float v2f __attribute__((ext_vector_type(2)));
typedef float v8f __attribute__((ext_vector_type(8)));
typedef int   v4i __attribute__((vector_size(16)));   // matches builtin param

#define WMMA_F32_16x16x4(A_, B_, C_) \
  __builtin_amdgcn_wmma_f32_16x16x4_f32(false, (A_), false, (B_), (short)0, (C_), false, false)

// global_load_async_to_lds_b128 wants (v4i AS1*, v4i AS3*, imm off, imm cpol)
#define AS1_V4I(p) ((__attribute__((address_space(1))) v4i*)(uintptr_t)(p))
#define AS3_V4I(p) ((__attribute__((address_space(3))) v4i*)(unsigned)(uintptr_t)(p))

namespace {
constexpr int kB  = 4;
constexpr int kH  = 16;
constexpr int kL  = 1024;
constexpr int kD  = 64;
constexpr int kDC = 128;           // concatenated feature width
constexpr int kBH = kB * kH;
constexpr float kEps = 1e-6f;
}  // namespace

// ---------------------------------------------------------------------------
// Phase 1: feature maps.
//   f1 = relu(x @ W1), f2 = sigmoid(x @ W2) * f1
//   XC[:, 0:64]  = f1
//   XC[:, 64:128]= f2  (Q side)   or   -alpha * f2  (K side)
// ---------------------------------------------------------------------------
__global__ __launch_bounds__(32)
void dla_fmap_kernel(const float* __restrict__ x,
                     const float* __restrict__ W1,
                     const float* __restrict__ W2,
                     const float* __restrict__ alpha,
                     float* __restrict__ XC,
                     int negAlphaSide) {
  const int bh   = blockIdx.x;     // b*H + h
  const int it   = blockIdx.y;     // 16-row tile index
  const int h    = bh % kH;
  const int lane = threadIdx.x;
  const int lo   = lane & 15;      // A: row M; B/C: col N
  const int hi   = lane >> 4;      // high half selects K (A/B) / M+8 (C)

  // A fragments of X tile: A[M][K], M = lo, K = 4*kc + 2*hi + {0,1}
  const float* X = x + ((size_t)bh * kL + (size_t)it * 16) * kD;
  v2f a[16];
#pragma unroll
  for (int kc = 0; kc < 16; ++kc) {
    const float* p = X + lo * kD + 4 * kc + 2 * hi;
    a[kc] = (v2f){p[0], p[1]};
  }

  const float* W1h = W1 + (size_t)h * kD * kD;
  const float* W2h = W2 + (size_t)h * kD * kD;
  const float scale2 = negAlphaSide ? -alpha[0] : 1.0f;
  float* XCrow = XC + ((size_t)bh * kL + (size_t)it * 16) * kDC;

#pragma unroll
  for (int nt = 0; nt < 4; ++nt) {           // 16-col output tiles
    v8f acc1 = (v8f){0.f, 0.f, 0.f, 0.f, 0.f, 0.f, 0.f, 0.f};
    v8f acc2 = (v8f){0.f, 0.f, 0.f, 0.f, 0.f, 0.f, 0.f, 0.f};
#pragma unroll
    for (int kc = 0; kc < 16; ++kc) {        // K = 64 in steps of 4
      const int kb = 4 * kc + 2 * hi;        // B[K][N] = W[d=K][e=N]
      v2f b1 = (v2f){W1h[(size_t)kb * kD + nt * 16 + lo],
                     W1h[(size_t)(kb + 1) * kD + nt * 16 + lo]};
      v2f b2 = (v2f){W2h[(size_t)kb * kD + nt * 16 + lo],
                     W2h[(size_t)(kb + 1) * kD + nt * 16 + lo]};
      acc1 = WMMA_F32_16x16x4(a[kc], b1, acc1);
      acc2 = WMMA_F32_16x16x4(a[kc], b2, acc2);
    }
    // C/D layout: element r -> (M = r + 8*hi, N = lo)
#pragma unroll
    for (int r = 0; r < 8; ++r) {
      const int M = r + 8 * hi;
      const float z1 = acc1[r];
      const float z2 = acc2[r];
      const float f1 = z1 > 0.f ? z1 : 0.f;            // relu
      const float g  = 1.f / (1.f + __expf(-z2));      // sigmoid
      XCrow[(size_t)M * kDC + nt * 16 + lo]      = f1;
      XCrow[(size_t)M * kDC + kD + nt * 16 + lo] = scale2 * (g * f1);
    }
  }
}

// ---------------------------------------------------------------------------
// Phase 2: causal differential attention with deferred normalization.
// 4 waves per block; KC/V column tiles double-buffered in LDS via async copy.
// ---------------------------------------------------------------------------
__global__ __launch_bounds__(128)
void dla_attn_kernel(const float* __restrict__ QC,
                     const float* __restrict__ KC,
                     const float* __restrict__ V,
                     float* __restrict__ out) {
  __shared__ __align__(16) float kcT[2][16 * kDC];   // 2 x 8 KB
  __shared__ __align__(16) float vT[2][16 * kD];     // 2 x 4 KB
  __shared__ float sT[4][16 * 18];                   // per-wave transpose tile

  const int bh    = blockIdx.x;
  const int ig    = blockIdx.y;          // group of 4 row tiles
  const int tid   = threadIdx.x;
  const int w     = tid >> 5;            // wave id 0..3
  const int lane  = tid & 31;
  const int lo    = lane & 15;
  const int hi    = lane >> 4;
  const int it    = ig * 4 + w;          // this wave's row tile
  const int itmax = ig * 4 + 3;

  const float* QCb = QC + (size_t)bh * kL * kDC;
  const float* KCb = KC + (size_t)bh * kL * kDC;
  const float* Vb  = V  + (size_t)bh * kL * kD;

  // Per-wave A fragments of its QC row tile: 16 rows x 128 features.
  v2f a[32];
#pragma unroll
  for (int kc = 0; kc < 32; ++kc) {
    const float* p = QCb + ((size_t)it * 16 + lo) * kDC + 4 * kc + 2 * hi;
    a[kc] = (v2f){p[0], p[1]};
  }

  // Cooperative async stage of one (KC, V) column tile into LDS buffer `buf`.
  // Both tiles are contiguous in memory: KC tile = 8 KB, V tile = 4 KB.
  auto stage_tiles = [&](int j, int buf) {
    const float* kcsrc = KCb + (size_t)j * 16 * kDC;
    const float* vsrc  = Vb + (size_t)j * 16 * kD;
#pragma unroll
    for (int c = 0; c < 4; ++c) {        // 512 x 16B chunks / 128 threads
      const int idx = tid + c * 128;
      __builtin_amdgcn_global_load_async_to_lds_b128(
          AS1_V4I(kcsrc + idx * 4), AS3_V4I(&kcT[buf][idx * 4]), 0, 0);
    }
#pragma unroll
    for (int c = 0; c < 2; ++c) {        // 256 x 16B chunks / 128 threads
      const int idx = tid + c * 128;
      __builtin_amdgcn_global_load_async_to_lds_b128(
          AS1_V4I(vsrc + idx * 4), AS3_V4I(&vT[buf][idx * 4]), 0, 0);
    }
  };

  v8f rsum = (v8f){0.f, 0.f, 0.f, 0.f, 0.f, 0.f, 0.f, 0.f};
  v8f oacc[4];
#pragma unroll
  for (int dt = 0; dt < 4; ++dt)
    oacc[dt] = (v8f){0.f, 0.f, 0.f, 0.f, 0.f, 0.f, 0.f, 0.f};

  stage_tiles(0, 0);

  for (int j = 0; j <= itmax; ++j) {     // block-uniform causal sweep
    const int buf = j & 1;
    __builtin_amdgcn_s_wait_asynccnt(0); // my async writes into buf landed
    __syncthreads();                     // everyone's writes landed

    if (j < itmax) stage_tiles(j + 1, buf ^ 1);  // prefetch behind compute

    if (j <= it) {                       // wave-uniform skip above diagonal
      const float* kc = kcT[buf];
      const float* vv = vT[buf];

      // --- scores: s = QC_tile (16x128) @ KC_tile^T, K=128 WMMA chain ---
      v8f s = (v8f){0.f, 0.f, 0.f, 0.f, 0.f, 0.f, 0.f, 0.f};
#pragma unroll
      for (int kci = 0; kci < 32; ++kci) {
        // B[K][N] = KC[m = N][feature = K]
        const float* p = kc + lo * kDC + 4 * kci + 2 * hi;
        v2f b = (v2f){p[0], p[1]};
        s = WMMA_F32_16x16x4(a[kci], b, s);
      }

      // --- diagonal causal mask (branch-free, EXEC preserved) ---
      if (j == it) {
#pragma unroll
        for (int r = 0; r < 8; ++r) {
          const int M = r + 8 * hi;      // keep col N <= row M
          s[r] = (lo <= M) ? s[r] : 0.f;
        }
      }

      // --- denominator partials ---
#pragma unroll
      for (int r = 0; r < 8; ++r) rsum[r] += s[r];

      // --- transpose C-layout -> A-layout through per-wave LDS tile ---
#pragma unroll
      for (int r = 0; r < 8; ++r) {
        const int M = r + 8 * hi;
        sT[w][M * 18 + lo] = s[r];
      }
      asm volatile("" ::: "memory");     // in-wave LDS ops are in-order

      // --- oacc += s_tile (16x16) @ V_tile (16x64), K=16 chain ---
#pragma unroll
      for (int c = 0; c < 4; ++c) {
        const int kb = 4 * c + 2 * hi;
        v2f as = (v2f){sT[w][lo * 18 + kb], sT[w][lo * 18 + kb + 1]};
        const float* vp = vv + (kb)*kD + lo;   // B[K][N] = V[m][d]
#pragma unroll
        for (int dt = 0; dt < 4; ++dt) {
          v2f bv = (v2f){vp[dt * 16], vp[kD + dt * 16]};
          oacc[dt] = WMMA_F32_16x16x4(as, bv, oacc[dt]);
        }
      }
    }
    __syncthreads();                     // readers done before next overwrite
  }

  // --- reduce row sums across the 16-lane halves (xor stays in-half) ---
#pragma unroll
  for (int r = 0; r < 8; ++r) {
    float v0 = rsum[r];
    v0 += __shfl_xor(v0, 1, 32);
    v0 += __shfl_xor(v0, 2, 32);
    v0 += __shfl_xor(v0, 4, 32);
    v0 += __shfl_xor(v0, 8, 32);
    rsum[r] = v0 + kEps;
  }

  // --- normalize + store: out[M][dt*16+lo] = oacc / denom(row M) ---
  float* ob = out + (size_t)bh * kL * kD + (size_t)it * 16 * kD;
#pragma unroll
  for (int dt = 0; dt < 4; ++dt) {
#pragma unroll
    for (int r = 0; r < 8; ++r) {
      const int M = r + 8 * hi;
      ob[(size_t)M * kD + dt * 16 + lo] = oacc[dt][r] / rsum[r];
    }
  }
}

// ---------------------------------------------------------------------------
extern "C" void kernel_launch(void* const* d_in, const int* in_sizes, int n_in,
                              void* d_out, int out_size, void* d_ws, size_t ws_size,
                              hipStream_t stream) {
  const float* q     = (const float*)d_in[0];
  const float* k     = (const float*)d_in[1];
  const float* v     = (const float*)d_in[2];
  const float* W1q   = (const float*)d_in[3];
  const float* W1k   = (const float*)d_in[4];
  const float* W2q   = (const float*)d_in[5];
  const float* W2k   = (const float*)d_in[6];
  const float* alpha = (const float*)d_in[7];
  float* out = (float*)d_out;

  float* QC = (float*)d_ws;                       // [BH, L, 128]
  float* KC = QC + (size_t)kBH * kL * kDC;        // [BH, L, 128]

  const dim3 gridF(kBH, kL / 16);
  dla_fmap_kernel<<<gridF, 32, 0, stream>>>(q, W1q, W2q, alpha, QC, 0);
  dla_fmap_kernel<<<gridF, 32, 0, stream>>>(k, W1k, W2k, alpha, KC, 1);

  const dim3 gridA(kBH, kL / 64);                 // 4 row tiles per block
  dla_attn_kernel<<<gridA, 128, 0, stream>>>(QC, KC, v, out);
}